// HTSK_87617332838941
// MI455X (gfx1250) — compile-verified
//
#include <hip/hip_runtime.h>

// HTSK fused kernel for MI455X (gfx1250, wave32, WMMA).
//
// out[b,o] = sum_r frs[b,r] * ( sum_i X[b,i]*W[r*D+i,o] + W[R*D+r,o] ) + bias[o]
// Implemented as one M=4096, N=64, K=32896 GEMM on V_WMMA_F32_16X16X4_F32,
// with the A matrix (frs[b,r]*X[b,i]) synthesized in-register from LDS tiles.

typedef __attribute__((ext_vector_type(2))) float v2f;
typedef __attribute__((ext_vector_type(8))) float v8f;

#define BDIM 4096
#define DDIM 256
#define RDIM 128
#define ODIM 64
#define BM   32      // rows of output per block
#define XS   260     // LDS row stride for X tile (conflict-free b64 loads)
#define FS   130     // LDS row stride for frs tile

__global__ __launch_bounds__(256)
void htsk_fused_kernel(const float* __restrict__ X,
                       const float* __restrict__ centers,
                       const float* __restrict__ sigmas,
                       const float* __restrict__ W,
                       const float* __restrict__ bias,
                       float* __restrict__ out)
{
    __shared__ float sX[BM * XS];   // 33280 B
    __shared__ float sF[BM * FS];   // 16640 B (logits, then softmaxed frs)

    const int tid  = threadIdx.x;
    const int row0 = blockIdx.x * BM;

    // ---- Phase 1a: stage X tile (32 x 256) into LDS, coalesced ----
    for (int idx = tid; idx < BM * DDIM; idx += 256) {
        int r = idx >> 8;          // row in tile
        int c = idx & (DDIM - 1);  // column
        sX[r * XS + c] = X[(size_t)(row0 + r) * DDIM + c];
    }
    __syncthreads();

    // ---- Phase 1b: logits[row][r] = mean_i -(x-c)^2 * (0.5/sigma^2 + eps) ----
    for (int p = tid; p < BM * RDIM; p += 256) {
        int row = p >> 7;           // 0..31
        int r   = p & (RDIM - 1);   // 0..127 (fastest across lanes -> coalesced)
        const float* xr = &sX[row * XS];
        float acc = 0.0f;
        for (int i = 0; i < DDIM; ++i) {
            float c  = centers[i * RDIM + r];
            float sg = sigmas[i * RDIM + r];
            float sc = 0.5f / (sg * sg) + 1e-8f;
            float dd = xr[i] - c;                 // xr[i] is an LDS broadcast
            acc = fmaf(-dd * dd, sc, acc);
        }
        sF[row * FS + r] = acc * (1.0f / (float)DDIM);
    }
    __syncthreads();

    // ---- Phase 1c: softmax over r, one thread per row (32 rows) ----
    if (tid < BM) {
        float* f = &sF[tid * FS];
        float m = -3.402823e38f;
        for (int r = 0; r < RDIM; ++r) m = fmaxf(m, f[r]);
        float s = 0.0f;
        for (int r = 0; r < RDIM; ++r) { float e = expf(f[r] - m); f[r] = e; s += e; }
        float inv = 1.0f / s;
        for (int r = 0; r < RDIM; ++r) f[r] *= inv;
    }
    __syncthreads();

    // ---- Phase 2: WMMA GEMM, 8 waves = 2 (M) x 4 (N) tiles of 16x16 ----
    const int lane   = tid & 31;
    const int wave   = tid >> 5;
    const int m_base = (wave >> 2) << 4;      // 0 or 16
    const int n_base = (wave & 3) << 4;       // 0,16,32,48
    const int col    = n_base + (lane & 15);  // output column (N)
    const int koff   = (lane >> 4) << 1;      // 0 (lanes 0-15) / 2 (lanes 16-31)
    const int arow   = m_base + (lane & 15);  // A-matrix row within tile

    const float* sXr = &sX[arow * XS + koff];
    const float* sFr = &sF[arow * FS];

    v8f acc = {};

    // Order-1 part: K = R*D, A[b, r*D+i] = frs[b,r]*X[b,i]
    for (int r = 0; r < RDIM; ++r) {
        float f = sFr[r];
        const float* wp = W + ((size_t)r * DDIM + koff) * ODIM + col;
        #pragma unroll 4
        for (int kk = 0; kk < DDIM; kk += 4) {
            v2f xa = *(const v2f*)(sXr + kk);   // ds_load_b64, conflict-free
            v2f a;  a.x = xa.x * f;  a.y = xa.y * f;
            v2f b;  b.x = wp[0];     b.y = wp[ODIM];
            acc = __builtin_amdgcn_wmma_f32_16x16x4_f32(
                      false, a, false, b, (short)0, acc, false, false);
            wp += 4 * ODIM;
        }
    }

    // Order-0 tail: K rows [R*D, R*D+R), A[b, R*D+r] = frs[b,r]
    {
        const float* wl = W + ((size_t)RDIM * DDIM + koff) * ODIM + col;
        #pragma unroll 4
        for (int kk = 0; kk < RDIM; kk += 4) {
            v2f a = *(const v2f*)(sFr + koff + kk);  // frs pair as A fragment
            v2f b;  b.x = wl[0];  b.y = wl[ODIM];
            acc = __builtin_amdgcn_wmma_f32_16x16x4_f32(
                      false, a, false, b, (short)0, acc, false, false);
            wl += 4 * ODIM;
        }
    }

    // ---- Epilogue: bias + store (C layout: VGPR p -> M=p / M=p+8) ----
    float bn = bias[col];
    const int mrow0 = row0 + m_base + ((lane >> 4) << 3);
    #pragma unroll
    for (int p = 0; p < 8; ++p) {
        out[(size_t)(mrow0 + p) * ODIM + col] = acc[p] + bn;
    }
}

extern "C" void kernel_launch(void* const* d_in, const int* in_sizes, int n_in,
                              void* d_out, int out_size, void* d_ws, size_t ws_size,
                              hipStream_t stream) {
    (void)in_sizes; (void)n_in; (void)d_ws; (void)ws_size; (void)out_size;
    const float* X       = (const float*)d_in[0];
    const float* centers = (const float*)d_in[1];
    const float* sigmas  = (const float*)d_in[2];
    const float* W       = (const float*)d_in[3];
    const float* b       = (const float*)d_in[4];
    float* out = (float*)d_out;

    htsk_fused_kernel<<<BDIM / BM, 256, 0, stream>>>(X, centers, sigmas, W, b, out);
}